// RelationNet_256_63196148793921
// MI455X (gfx1250) — compile-verified
//
#include <hip/hip_runtime.h>
#include <hip/hip_bf16.h>

// ---------------------------------------------------------------------------
// RelationNet (cov-metric) pipeline for MI455X / gfx1250, wave32, f32 WMMA.
// All GEMM-shaped phases use V_WMMA_F32_16X16X4_F32 (full f32 precision —
// matches the f32 reference; at ~80 GFLOP total and with the whole working
// set L2-resident (192 MB), precision costs nothing here).
// Workspace layout (floats):
//   A region : 11,289,600  (activations ping:  x0 / act1 / act2 / act3 / act4)
//   B region : 45,158,400  (raw conv pong; max = conv1 raw 100x64x84x84)
//   COV      :    327,680  (5 x 256 x 256)
//   MEAN/RSTD:      4,096
// ---------------------------------------------------------------------------

typedef float v2f __attribute__((ext_vector_type(2)));
typedef float v8f __attribute__((ext_vector_type(8)));

#define NQ      75
#define NCLS    5
#define BS      5
#define NTOT    100
#define CFEAT   256
#define PPOS    441      // 21*21
#define NSUP    2205     // BS*PPOS
#define NEG_SLOPE 0.2f
#define BN_EPS  1e-5f

__device__ __forceinline__ float lrelu(float v) {
  return v >= 0.0f ? v : NEG_SLOPE * v;
}

__device__ __forceinline__ int bn_group(int img) {
  return img < NQ ? 0 : 1 + (img - NQ) / BS;
}

// ---------------------------------------------------------------------------
// Implicit-GEMM 3x3 SAME conv, stride 1, via f32 WMMA 16x16x4.
// Compile-time <CIN, COUT, S>: all div/mod fold to mul-shift; K loop unrolls.
// M = COUT, K = CIN*9, N = spatial positions; one wave = 16x16 output tile.
// Fragment layouts per ISA 7.12.2: A/B lane L holds (row/col = L&15,
// k = kb + 2*(L>>4) + j) in VGPR j; C/D VGPR r = (m = r + 8*(L>>4), n = L&15).
// ---------------------------------------------------------------------------
template <int CIN, int COUT, int S>
__global__ __launch_bounds__(256) void conv_wmma_kernel(
    const float* __restrict__ in, const float* __restrict__ w,
    float* __restrict__ raw)
{
  constexpr int Ktot = CIN * 9;
  constexpr int SS = S * S;
  constexpr int PT = (SS + 15) / 16;
  constexpr int tilesM = COUT / 16;
  constexpr int totalWaves = NTOT * tilesM * PT;

  const int wid = (int)((blockIdx.x * blockDim.x + threadIdx.x) >> 5);
  if (wid >= totalWaves) return;

  const int lane = threadIdx.x & 31;
  const int half = lane >> 4;      // 0: K pair {0,1}, 1: K pair {2,3}
  const int ln   = lane & 15;

  const int mt  = wid % tilesM;
  const int t1  = wid / tilesM;
  const int pt  = t1 % PT;
  const int img = t1 / PT;

  const int co = mt * 16 + ln;           // A row (output channel)
  const int p  = pt * 16 + ln;           // B col (spatial position)
  const bool pv = p < SS;
  const int y = pv ? p / S : 0;
  const int x = pv ? p - y * S : 0;

  const float* __restrict__ in_img = in + (size_t)img * CIN * SS;
  const float* __restrict__ wrow   = w + (size_t)co * Ktot;

  v8f acc = {0.f, 0.f, 0.f, 0.f, 0.f, 0.f, 0.f, 0.f};

#pragma unroll 4
  for (int kb = 0; kb < Ktot; kb += 4) {
    const int k0 = kb + 2 * half;
    const int k1 = k0 + 1;
    v2f a, b;
    a.x = (k0 < Ktot) ? wrow[k0] : 0.f;
    a.y = (k1 < Ktot) ? wrow[k1] : 0.f;

    float b0 = 0.f, b1 = 0.f;
    if (pv) {
      if (k0 < Ktot) {
        int ci = k0 / 9, r = k0 - ci * 9;
        int ky = r / 3, kx = r - ky * 3;
        int iy = y + ky - 1, ix = x + kx - 1;
        if (iy >= 0 && iy < S && ix >= 0 && ix < S)
          b0 = in_img[(ci * S + iy) * S + ix];
      }
      if (k1 < Ktot) {
        int ci = k1 / 9, r = k1 - ci * 9;
        int ky = r / 3, kx = r - ky * 3;
        int iy = y + ky - 1, ix = x + kx - 1;
        if (iy >= 0 && iy < S && ix >= 0 && ix < S)
          b1 = in_img[(ci * S + iy) * S + ix];
      }
    }
    b.x = b0; b.y = b1;

    acc = __builtin_amdgcn_wmma_f32_16x16x4_f32(
        false, a, false, b, (short)0, acc, false, false);
  }

  if (pv) {
#pragma unroll
    for (int r = 0; r < 8; ++r) {
      const int m = r + 8 * half;
      raw[((size_t)img * COUT + mt * 16 + m) * SS + p] = acc[r];
    }
  }
}

// ---------------------------------------------------------------------------
// Per-(BN-group, channel) batch stats: biased variance (matches jnp.var).
// Group 0 = 75 query images; groups 1..5 = each support class (5 images).
// Templated on <COUT, S>: img-outer / strided-position-inner loops — zero
// div/mod in the hot loop, fully coalesced reads.
// ---------------------------------------------------------------------------
template <int COUT, int S>
__global__ __launch_bounds__(256) void bn_stats_kernel(
    const float* __restrict__ raw, float* __restrict__ mean,
    float* __restrict__ rstd)
{
  constexpr int SS = S * S;
  const int g = blockIdx.x / COUT;
  const int c = blockIdx.x % COUT;
  const int i0 = (g == 0) ? 0 : NQ + (g - 1) * BS;
  const int nImg = (g == 0) ? NQ : BS;

  float s = 0.f, q = 0.f;
  for (int img = i0; img < i0 + nImg; ++img) {
    const float* __restrict__ base = raw + ((size_t)img * COUT + c) * SS;
    for (int pp = threadIdx.x; pp < SS; pp += 256) {
      const float v = base[pp];
      s += v; q += v * v;
    }
  }
  __shared__ float ssum[256], ssq[256];
  ssum[threadIdx.x] = s; ssq[threadIdx.x] = q;
  __syncthreads();
  for (int o = 128; o > 0; o >>= 1) {
    if (threadIdx.x < o) {
      ssum[threadIdx.x] += ssum[threadIdx.x + o];
      ssq[threadIdx.x]  += ssq[threadIdx.x + o];
    }
    __syncthreads();
  }
  if (threadIdx.x == 0) {
    const float cnt = (float)(nImg * SS);
    const float m = ssum[0] / cnt;
    const float var = ssq[0] / cnt - m * m;
    mean[g * COUT + c] = m;
    rstd[g * COUT + c] = rsqrtf(var + BN_EPS);
  }
}

// ---------------------------------------------------------------------------
// Fused BN(affine) + LeakyReLU + optional 2x2 maxpool (stride 2).
// Templated: all index decode divisors are compile-time constants.
// ---------------------------------------------------------------------------
template <int COUT, int S, int POOL>
__global__ __launch_bounds__(256) void bn_lrelu_kernel(
    const float* __restrict__ raw, const float* __restrict__ mean,
    const float* __restrict__ rstd, const float* __restrict__ gamma,
    const float* __restrict__ beta, float* __restrict__ act)
{
  constexpr int So = POOL ? (S / 2) : S;
  constexpr int SoSo = So * So;
  constexpr long long total = (long long)NTOT * COUT * SoSo;
  const long long idx = (long long)blockIdx.x * blockDim.x + threadIdx.x;
  if (idx >= total) return;

  const int sp  = (int)(idx % SoSo);
  const int c   = (int)((idx / SoSo) % COUT);
  const int img = (int)(idx / ((long long)SoSo * COUT));
  const int g = bn_group(img);

  const float m = mean[g * COUT + c];
  const float r = rstd[g * COUT + c];
  const float sc = gamma[c] * r;
  const float sh = beta[c] - m * sc;

  const float* __restrict__ rb = raw + ((size_t)img * COUT + c) * S * S;
  float o;
  if (POOL) {
    const int oy = sp / So, ox = sp % So;
    float best = -3.402823466e38f;
#pragma unroll
    for (int dy = 0; dy < 2; ++dy)
#pragma unroll
      for (int dx = 0; dx < 2; ++dx) {
        const float v = lrelu(rb[(2 * oy + dy) * S + (2 * ox + dx)] * sc + sh);
        best = fmaxf(best, v);
      }
    o = best;
  } else {
    o = lrelu(rb[sp] * sc + sh);
  }
  act[((size_t)img * COUT + c) * SoSo + sp] = o;
}

// ---------------------------------------------------------------------------
// In-place centering: query per (b,c) over 441; support per (k,c) over 2205.
// ---------------------------------------------------------------------------
__global__ __launch_bounds__(128) void center_query_kernel(float* __restrict__ act)
{
  const int b = blockIdx.x / CFEAT;
  const int c = blockIdx.x % CFEAT;
  float* row = act + ((size_t)b * CFEAT + c) * PPOS;
  float s = 0.f;
  for (int i = threadIdx.x; i < PPOS; i += 128) s += row[i];
  __shared__ float sh[128];
  sh[threadIdx.x] = s; __syncthreads();
  for (int o = 64; o > 0; o >>= 1) {
    if (threadIdx.x < o) sh[threadIdx.x] += sh[threadIdx.x + o];
    __syncthreads();
  }
  const float mu = sh[0] / (float)PPOS;
  for (int i = threadIdx.x; i < PPOS; i += 128) row[i] -= mu;
}

__global__ __launch_bounds__(128) void center_support_kernel(float* __restrict__ act)
{
  const int k = blockIdx.x / CFEAT;
  const int c = blockIdx.x % CFEAT;
  float s = 0.f;
  for (int bs = 0; bs < BS; ++bs) {
    const float* row = act + ((size_t)(NQ + k * BS + bs) * CFEAT + c) * PPOS;
    for (int i = threadIdx.x; i < PPOS; i += 128) s += row[i];
  }
  __shared__ float sh[128];
  sh[threadIdx.x] = s; __syncthreads();
  for (int o = 64; o > 0; o >>= 1) {
    if (threadIdx.x < o) sh[threadIdx.x] += sh[threadIdx.x + o];
    __syncthreads();
  }
  const float mu = sh[0] / (float)NSUP;
  for (int bs = 0; bs < BS; ++bs) {
    float* row = act + ((size_t)(NQ + k * BS + bs) * CFEAT + c) * PPOS;
    for (int i = threadIdx.x; i < PPOS; i += 128) row[i] -= mu;
  }
}

// ---------------------------------------------------------------------------
// Per-class covariance: cov[k] = sf sf^T / (NSUP-1), sf = centered [256, 2205]
// (column n maps to support image bs = n/441, position n%441).
// One wave per 16x16 output tile; 5*16*16 = 1280 waves; K padded to 2208.
// ---------------------------------------------------------------------------
__global__ __launch_bounds__(256) void cov_wmma_kernel(
    const float* __restrict__ actS, float* __restrict__ cov)
{
  const int wid = (int)((blockIdx.x * blockDim.x + threadIdx.x) >> 5);
  if (wid >= NCLS * 16 * 16) return;
  const int lane = threadIdx.x & 31;
  const int half = lane >> 4, ln = lane & 15;

  const int mt = wid % 16;
  const int nt = (wid / 16) % 16;
  const int kc = wid / 256;

  auto sf = [&](int c, int n) -> float {
    if (n >= NSUP) return 0.f;
    const int bs = n / PPOS, pp = n - bs * PPOS;
    return actS[(((size_t)(NQ + kc * BS + bs)) * CFEAT + c) * PPOS + pp];
  };

  const int c = mt * 16 + ln;
  const int d = nt * 16 + ln;
  v8f acc = {0.f, 0.f, 0.f, 0.f, 0.f, 0.f, 0.f, 0.f};

#pragma unroll 8
  for (int kb = 0; kb < 2208; kb += 4) {
    const int kk = kb + 2 * half;
    v2f a, b;
    a.x = sf(c, kk);     a.y = sf(c, kk + 1);
    b.x = sf(d, kk);     b.y = sf(d, kk + 1);
    acc = __builtin_amdgcn_wmma_f32_16x16x4_f32(
        false, a, false, b, (short)0, acc, false, false);
  }

  const float inv = 1.0f / (float)(NSUP - 1);
#pragma unroll
  for (int r = 0; r < 8; ++r) {
    const int cc = mt * 16 + r + 8 * half;
    cov[((size_t)kc * CFEAT + cc) * CFEAT + nt * 16 + ln] = acc[r] * inv;
  }
}

// ---------------------------------------------------------------------------
// Fused Mahalanobis + LeakyReLU + Conv1d(441) classifier.
// maha[b,k,p] = sum_{c,d} qf[b,c,p] cov[k,c,d] qf[b,d,p]  (cov symmetric)
// One wave per (b, k, 16-pos tile), one wave per block: the 256x16 q tile is
// staged ONCE into a 16 KB LDS slice (coalesced global reads), then all 16
// d-block GEMMs and the second q contraction read it via conflict-free DS
// loads — removes the 16x redundant global re-fetch of the q tile.
// ---------------------------------------------------------------------------
__global__ __launch_bounds__(32) void maha_wmma_kernel(
    const float* __restrict__ actQ, const float* __restrict__ cov,
    const float* __restrict__ wc, float* __restrict__ out)
{
  constexpr int PT = (PPOS + 15) / 16;   // 28
  __shared__ float qtile[CFEAT * 16];    // [row kk][col j], 16 KB

  const int wid = blockIdx.x;
  if (wid >= NQ * NCLS * PT) return;
  const int lane = threadIdx.x & 31;
  const int half = lane >> 4, ln = lane & 15;

  const int pt = wid % PT;
  const int k  = (wid / PT) % NCLS;
  const int b  = wid / (PT * NCLS);

  const float* __restrict__ qbase = actQ + (size_t)b * CFEAT * PPOS;
  const float* __restrict__ cbase = cov + (size_t)k * CFEAT * CFEAT;

  // stage q tile: qtile[row*16 + col] = qf[b, row, pt*16+col] (0 if OOB)
  for (int i = lane; i < CFEAT * 16; i += 32) {
    const int row = i >> 4, col = i & 15;
    const int pp = pt * 16 + col;
    qtile[i] = (pp < PPOS) ? qbase[(size_t)row * PPOS + pp] : 0.f;
  }
  __builtin_amdgcn_wave_barrier();   // same-wave DS ops are in-order (ISA 5.7)

  const int p = pt * 16 + ln;
  const bool pv = p < PPOS;

  float msum = 0.f;
  for (int db = 0; db < 16; ++db) {
    v8f acc = {0.f, 0.f, 0.f, 0.f, 0.f, 0.f, 0.f, 0.f};
    const float* __restrict__ crow = cbase + (size_t)(db * 16 + ln) * CFEAT;
#pragma unroll 8
    for (int kb = 0; kb < CFEAT; kb += 4) {
      const int kk = kb + 2 * half;
      v2f a, bfr;
      a.x = crow[kk];
      a.y = crow[kk + 1];
      bfr.x = qtile[kk * 16 + ln];
      bfr.y = qtile[(kk + 1) * 16 + ln];
      acc = __builtin_amdgcn_wmma_f32_16x16x4_f32(
          false, a, false, bfr, (short)0, acc, false, false);
    }
#pragma unroll
    for (int r = 0; r < 8; ++r) {
      const int d = db * 16 + r + 8 * half;
      msum += acc[r] * qtile[d * 16 + ln];   // OOB cols are zero in the tile
    }
  }
  // combine the two half-wave row groups (lanes L and L+16 share column p)
  msum += __shfl_xor(msum, 16, 32);

  const float y = lrelu(msum);
  float contrib = (pv && half == 0) ? y * wc[p] : 0.f;
  // reduce the 16 distinct p-lanes (upper half zeroed)
  for (int o = 8; o >= 1; o >>= 1) contrib += __shfl_xor(contrib, o, 32);
  if (lane == 0) atomicAdd(&out[b * NCLS + k], contrib);
}

__global__ void zero_kernel(float* __restrict__ p, int n)
{
  const int i = blockIdx.x * blockDim.x + threadIdx.x;
  if (i < n) p[i] = 0.f;
}

// ---------------------------------------------------------------------------
extern "C" void kernel_launch(void* const* d_in, const int* in_sizes, int n_in,
                              void* d_out, int out_size, void* d_ws, size_t ws_size,
                              hipStream_t stream) {
  const float* in1 = (const float*)d_in[0];   // [75,3,84,84]
  const float* in2 = (const float*)d_in[1];   // [5,5,3,84,84]
  const float* w1 = (const float*)d_in[2];
  const float* g1 = (const float*)d_in[3];
  const float* b1 = (const float*)d_in[4];
  const float* w2 = (const float*)d_in[5];
  const float* g2 = (const float*)d_in[6];
  const float* b2 = (const float*)d_in[7];
  const float* w3 = (const float*)d_in[8];
  const float* g3 = (const float*)d_in[9];
  const float* b3 = (const float*)d_in[10];
  const float* w4 = (const float*)d_in[11];
  const float* g4 = (const float*)d_in[12];
  const float* b4 = (const float*)d_in[13];
  const float* wc = (const float*)d_in[14];
  float* out = (float*)d_out;

  const size_t A_ELEMS = 11289600;   // max activation (act1/act4)
  const size_t B_ELEMS = 45158400;   // max raw conv (conv1)
  const size_t COV_ELEMS = (size_t)NCLS * CFEAT * CFEAT;
  const size_t NEED = (A_ELEMS + B_ELEMS + COV_ELEMS + 4096) * sizeof(float);
  if (ws_size < NEED) return;

  float* ws = (float*)d_ws;
  float* A    = ws;
  float* B    = A + A_ELEMS;
  float* COV  = B + B_ELEMS;
  float* MEAN = COV + COV_ELEMS;
  float* RSTD = MEAN + 2048;

  // concat [query | support] -> x0 in region A
  hipMemcpyAsync(A, in1, (size_t)NQ * 3 * 84 * 84 * sizeof(float),
                 hipMemcpyDeviceToDevice, stream);
  hipMemcpyAsync(A + (size_t)NQ * 3 * 84 * 84, in2,
                 (size_t)25 * 3 * 84 * 84 * sizeof(float),
                 hipMemcpyDeviceToDevice, stream);

  auto conv_blocks = [](int Cout, int S) {
    const long long waves = (long long)NTOT * (Cout / 16) * ((S * S + 15) / 16);
    return (int)((waves * 32 + 255) / 256);
  };
  auto elem_blocks = [](int Cout, int So) {
    const long long tot = (long long)NTOT * Cout * So * So;
    return (int)((tot + 255) / 256);
  };

  // stage 1: [100,3,84,84] -> act1 [100,64,42,42]
  conv_wmma_kernel<3, 64, 84><<<conv_blocks(64, 84), 256, 0, stream>>>(A, w1, B);
  bn_stats_kernel<64, 84><<<(NCLS + 1) * 64, 256, 0, stream>>>(B, MEAN, RSTD);
  bn_lrelu_kernel<64, 84, 1><<<elem_blocks(64, 42), 256, 0, stream>>>(
      B, MEAN, RSTD, g1, b1, A);

  // stage 2: -> act2 [100,96,21,21]
  conv_wmma_kernel<64, 96, 42><<<conv_blocks(96, 42), 256, 0, stream>>>(A, w2, B);
  bn_stats_kernel<96, 42><<<(NCLS + 1) * 96, 256, 0, stream>>>(B, MEAN, RSTD);
  bn_lrelu_kernel<96, 42, 1><<<elem_blocks(96, 21), 256, 0, stream>>>(
      B, MEAN, RSTD, g2, b2, A);

  // stage 3: -> act3 [100,128,21,21]
  conv_wmma_kernel<96, 128, 21><<<conv_blocks(128, 21), 256, 0, stream>>>(A, w3, B);
  bn_stats_kernel<128, 21><<<(NCLS + 1) * 128, 256, 0, stream>>>(B, MEAN, RSTD);
  bn_lrelu_kernel<128, 21, 0><<<elem_blocks(128, 21), 256, 0, stream>>>(
      B, MEAN, RSTD, g3, b3, A);

  // stage 4: -> act4 [100,256,21,21]
  conv_wmma_kernel<128, 256, 21><<<conv_blocks(256, 21), 256, 0, stream>>>(A, w4, B);
  bn_stats_kernel<256, 21><<<(NCLS + 1) * 256, 256, 0, stream>>>(B, MEAN, RSTD);
  bn_lrelu_kernel<256, 21, 0><<<elem_blocks(256, 21), 256, 0, stream>>>(
      B, MEAN, RSTD, g4, b4, A);

  center_query_kernel<<<NQ * CFEAT, 128, 0, stream>>>(A);
  center_support_kernel<<<NCLS * CFEAT, 128, 0, stream>>>(A);

  {
    const long long waves = (long long)NCLS * 16 * 16;
    cov_wmma_kernel<<<(int)((waves * 32 + 255) / 256), 256, 0, stream>>>(A, COV);
  }

  zero_kernel<<<(out_size + 255) / 256, 256, 0, stream>>>(out, out_size);

  maha_wmma_kernel<<<NQ * NCLS * ((PPOS + 15) / 16), 32, 0, stream>>>(
      A, COV, wc, out);
}